// YOLOLoss__19413252178336
// MI455X (gfx1250) — compile-verified
//
#include <hip/hip_runtime.h>
#include <hip/hip_bf16.h>

// ---- problem constants (from reference) ----
#define A_NUM   3
#define NCLS    80
#define CP5     85          // NCLS + 5
#define BDIM    64
#define GDIM    52
#define GG      (GDIM * GDIM)        // 2704
#define NTOT    (BDIM * GG)          // 173056 = 676 * 256 exactly

typedef __attribute__((ext_vector_type(2))) float v2f;
typedef __attribute__((ext_vector_type(8))) float v8f;

// Exact 32-lane sum using V_WMMA_F32_16X16X4_F32.
// A (16x4): lane L<16 -> (M=L,K=0)=a0,(M=L,K=1)=a1 ; lane L>=16 -> (M=L-16,K=2)=a0,(K=3)=a1.
// With a0=v, a1=0, B=ones: D[m][n] = v[m] + v[m+16] for all n.
// D layout: lane l<16 VGPR r holds D[r][l]; lane l>=16 holds D[8+r][l-16].
// So sum of 8 D regs gives half-sum per 16-lane half; xor-16 shuffle completes it.
__device__ __forceinline__ float wave_sum32(float v) {
    v2f a; a[0] = v;    a[1] = 0.0f;
    v2f b; b[0] = 1.0f; b[1] = 1.0f;
    v8f c = {};
    c = __builtin_amdgcn_wmma_f32_16x16x4_f32(
            /*neg_a=*/false, a, /*neg_b=*/false, b,
            /*c_mod=*/(short)0, c, /*reuse_a=*/false, /*reuse_b=*/false);
    float s = c[0] + c[1] + c[2] + c[3] + c[4] + c[5] + c[6] + c[7];
    s += __shfl_xor(s, 16, 32);
    return s;   // full 32-lane sum, broadcast to every lane
}

__global__ void yolo_init_acc(float* __restrict__ acc) {
    if (threadIdx.x < 3) acc[threadIdx.x] = 0.0f;
}

__global__ void __launch_bounds__(256)
yolo_loss_main(const float* __restrict__ pred,
               const float* __restrict__ tgt,
               float* __restrict__ acc) {
    __shared__ float sh[3];
    if (threadIdx.x == 0) { sh[0] = 0.0f; sh[1] = 0.0f; sh[2] = 0.0f; }
    __syncthreads();

    const int n = blockIdx.x * blockDim.x + threadIdx.x;

    float objAcc = 0.0f;    // 0.5*(sq_xy+sq_wh) - log(conf) - picked_logprob, obj cells
    float noAcc  = 0.0f;    // sum_a -log(1-conf_a), non-obj cells
    float cnt    = 0.0f;    // obj count

    if (n < NTOT) {
        const int gx = n % GDIM;
        const int t2 = n / GDIM;
        const int gy = t2 % GDIM;
        const int bb = t2 / GDIM;

        const size_t cell = (size_t)gy * GDIM + gx;
        const float* __restrict__ pb = pred + (size_t)bb * (A_NUM * CP5) * GG + cell;
        const float* __restrict__ tb = tgt  + (size_t)bb * 6 * GG + cell;

        // target: [x,y,w,h,obj,class], channel stride GG
        const float tx   = tb[0];
        const float ty   = tb[(size_t)1 * GG];
        const float tw   = tb[(size_t)2 * GG];
        const float th   = tb[(size_t)3 * GG];
        const float tobj = tb[(size_t)4 * GG];
        const float tcls = tb[(size_t)5 * GG];

        const float cellsz = 1.0f / (float)GDIM;
        const float offx = (float)gx * cellsz;   // x += col * cell
        const float offy = (float)gy * cellsz;   // y += row * cell

        float px[A_NUM], py[A_NUM], pw[A_NUM], ph[A_NUM], pc[A_NUM];
        #pragma unroll
        for (int a = 0; a < A_NUM; ++a) {
            const float* __restrict__ pa = pb + (size_t)(a * CP5) * GG;
            px[a] = pa[0]              + offx;
            py[a] = pa[(size_t)1 * GG] + offy;
            pw[a] = pa[(size_t)2 * GG];
            ph[a] = pa[(size_t)3 * GG];
            pc[a] = pa[(size_t)4 * GG];
        }

        // target corners + area
        const float bx1 = tx - tw * 0.5f, by1 = ty - th * 0.5f;
        const float bx2 = tx + tw * 0.5f, by2 = ty + th * 0.5f;
        const float areaB = tw * th;

        // best-anchor by IoU (first-max tie like jnp.argmax)
        int   best    = 0;
        float bestIou = -1.0f;
        #pragma unroll
        for (int a = 0; a < A_NUM; ++a) {
            const float ax1 = px[a] - pw[a] * 0.5f, ay1 = py[a] - ph[a] * 0.5f;
            const float ax2 = px[a] + pw[a] * 0.5f, ay2 = py[a] + ph[a] * 0.5f;
            float iw = fminf(ax2, bx2) - fmaxf(ax1, bx1); iw = fmaxf(iw, 0.0f);
            float ih = fminf(ay2, by2) - fmaxf(ay1, by1); ih = fmaxf(ih, 0.0f);
            const float inter = iw * ih;
            const float uni   = pw[a] * ph[a] + areaB - inter;
            const float iou   = inter / uni;
            if (iou > bestIou) { bestIou = iou; best = a; }
        }

        const bool isObj = (tobj == 1.0f);
        cnt = isObj ? 1.0f : 0.0f;

        if (isObj) {
            const float dx = px[best] - tx;
            const float dy = py[best] - ty;
            const float sqxy = dx * dx + dy * dy;
            const float dw = sqrtf(pw[best]) - sqrtf(tw);
            const float dh = sqrtf(ph[best]) - sqrtf(th);
            const float sqwh = dw * dw + dh * dh;

            // class cross-entropy: online logsumexp over 80 strided logits
            const int cls = (int)tcls;
            const float* __restrict__ pcls = pb + (size_t)(best * CP5 + 5) * GG;
            float m = -1e30f, s = 0.0f, picked = 0.0f;
            for (int c = 0; c < NCLS; ++c) {
                const float v = pcls[(size_t)c * GG];
                if (c + 8 < NCLS)
                    __builtin_prefetch(&pcls[(size_t)(c + 8) * GG], 0, 0);
                if (c == cls) picked = v;
                if (v > m) { s = s * __expf(m - v) + 1.0f; m = v; }
                else       { s += __expf(v - m); }
            }
            const float lse = m + __logf(s);
            objAcc = 0.5f * (sqxy + sqwh) - __logf(pc[best]) + (lse - picked);
        } else {
            noAcc = -(__logf(1.0f - pc[0]) +
                      __logf(1.0f - pc[1]) +
                      __logf(1.0f - pc[2]));
        }
    }

    // --- wave reduction via WMMA (all branches re-converged; EXEC all ones) ---
    const float rObj = wave_sum32(objAcc);
    const float rNo  = wave_sum32(noAcc);
    const float rCnt = wave_sum32(cnt);

    if ((threadIdx.x & 31) == 0) {
        atomicAdd(&sh[0], rObj);   // ds_add_f32
        atomicAdd(&sh[1], rNo);
        atomicAdd(&sh[2], rCnt);
    }
    __syncthreads();
    if (threadIdx.x == 0) {
        atomicAdd(&acc[0], sh[0]); // global_atomic_add_f32
        atomicAdd(&acc[1], sh[1]);
        atomicAdd(&acc[2], sh[2]);
    }
}

__global__ void yolo_finalize(const float* __restrict__ acc, float* __restrict__ out) {
    const float nObj  = acc[2];
    const float nNo   = (float)NTOT - nObj;
    out[0] = acc[0] / nObj + acc[1] / ((float)A_NUM * nNo);
}

extern "C" void kernel_launch(void* const* d_in, const int* in_sizes, int n_in,
                              void* d_out, int out_size, void* d_ws, size_t ws_size,
                              hipStream_t stream) {
    (void)in_sizes; (void)n_in; (void)out_size; (void)ws_size;
    const float* pred = (const float*)d_in[0];   // (64, 255, 52, 52) f32
    const float* tgt  = (const float*)d_in[1];   // (64, 6, 52, 52)  f32
    float* out = (float*)d_out;                  // scalar f32
    float* acc = (float*)d_ws;                   // 3 floats scratch

    yolo_init_acc<<<1, 32, 0, stream>>>(acc);
    yolo_loss_main<<<NTOT / 256, 256, 0, stream>>>(pred, tgt, acc);
    yolo_finalize<<<1, 1, 0, stream>>>(acc, out);
}